// ASGLoss_4801773436947
// MI455X (gfx1250) — compile-verified
//
#include <hip/hip_runtime.h>
#include <hip/hip_bf16.h>

#define NEGV  (-1e30f)
#define LOG2E (1.4426950408889634f)
#define LN2   (0.6931471805599453f)

typedef float v2f __attribute__((ext_vector_type(2)));
typedef float v8f __attribute__((ext_vector_type(8)));

static __device__ __forceinline__ float max8(const v8f& a) {
  // balanced tree, depth 3
  return fmaxf(fmaxf(fmaxf(a[0], a[1]), fmaxf(a[2], a[3])),
               fmaxf(fmaxf(a[4], a[5]), fmaxf(a[6], a[7])));
}

// ---------------------------------------------------------------------------
// Kernel 1: _fcc via fp32 WMMA in exp space, alpha kept in base-2 domain.
// One wave per 16 batch columns. Per step:
//   M_b  = colmax(beta)                       (balanced max tree + shfl_xor 16)
//   X    = 2^(beta - M)                       (raw v_exp) -> LDS Xl[b][m]
//   S    = E * X  (E = exp(trans), 48x48)     -> 3 x 12 WMMA f32 16x16x4
//   beta = x_t*log2e + M + log2(S)            (raw v_log + fma)
// E lives in registers in WMMA-A layout (72 VGPRs).
// ---------------------------------------------------------------------------
__launch_bounds__(32)
__global__ void asg_fcc_kernel(const float* __restrict__ x,     // [B,T,48]
                               const float* __restrict__ trans, // [48,48]
                               float* __restrict__ fcc,         // [B]
                               int T) {
  const int lane = threadIdx.x;            // 0..31
  const int col  = lane & 15;              // batch column within tile
  const int half = lane >> 4;              // 0 | 1
  const int roff = half * 8;               // D-layout row offset
  const int s0   = half * 2;               // A/B-layout K select
  const int b    = blockIdx.x * 16 + col;  // global batch index

  __shared__ float Xl[16 * 48];            // X[b][m], 3 KB, wave-private

  // ---- E = exp(trans) in WMMA A layout: a.x=E[n][4k+s0], a.y=E[n][4k+s0+1]
  v2f E[3][12];
#pragma unroll
  for (int nt = 0; nt < 3; ++nt) {
    const int n = nt * 16 + col;
#pragma unroll
    for (int kt = 0; kt < 12; ++kt) {
      const float* p = trans + n * 48 + kt * 4 + s0;
      v2f e;
      e.x = __builtin_amdgcn_exp2f(p[0] * LOG2E);
      e.y = __builtin_amdgcn_exp2f(p[1] * LOG2E);
      E[nt][kt] = e;
    }
  }

  // ---- beta0 = x[:,0] * log2e in D layout: al[nt][v] = beta[nt*16+v+roff][col]
  const float* xb = x + (size_t)b * T * 48;
  v8f al[3];
#pragma unroll
  for (int nt = 0; nt < 3; ++nt) {
    const float4 lo = *(const float4*)(xb + nt * 16 + roff);
    const float4 hi = *(const float4*)(xb + nt * 16 + roff + 4);
    al[nt][0] = lo.x * LOG2E; al[nt][1] = lo.y * LOG2E;
    al[nt][2] = lo.z * LOG2E; al[nt][3] = lo.w * LOG2E;
    al[nt][4] = hi.x * LOG2E; al[nt][5] = hi.y * LOG2E;
    al[nt][6] = hi.z * LOG2E; al[nt][7] = hi.w * LOG2E;
  }

  float*       XlS = Xl + col * 48;        // store base: this lane's column b
  const float* XlL = Xl + col * 48;        // B-operand load base (N = lane%16)

  for (int t = 1; t < T; ++t) {
    const float* xt = xb + (size_t)t * 48;
    __builtin_prefetch(xt + 48, 0, 1);     // next step's row -> global_prefetch_b8

    // column max over 48 rows (balanced tree + cross-half combine)
    float M = fmaxf(fmaxf(max8(al[0]), max8(al[1])), max8(al[2]));
    M = fmaxf(M, __shfl_xor(M, 16, 32));

    // X = 2^(beta - M) staged to LDS as Xl[b][m] (rows contiguous -> b128)
#pragma unroll
    for (int nt = 0; nt < 3; ++nt) {
      float4 lo, hi;
      lo.x = __builtin_amdgcn_exp2f(al[nt][0] - M);
      lo.y = __builtin_amdgcn_exp2f(al[nt][1] - M);
      lo.z = __builtin_amdgcn_exp2f(al[nt][2] - M);
      lo.w = __builtin_amdgcn_exp2f(al[nt][3] - M);
      hi.x = __builtin_amdgcn_exp2f(al[nt][4] - M);
      hi.y = __builtin_amdgcn_exp2f(al[nt][5] - M);
      hi.z = __builtin_amdgcn_exp2f(al[nt][6] - M);
      hi.w = __builtin_amdgcn_exp2f(al[nt][7] - M);
      *(float4*)(XlS + nt * 16 + roff)     = lo;
      *(float4*)(XlS + nt * 16 + roff + 4) = hi;
    }

    // B operand tiles: bt.x = X[4k+s0][col], bt.y = X[4k+s0+1][col]
    v2f Bt[12];
#pragma unroll
    for (int kt = 0; kt < 12; ++kt)
      Bt[kt] = *(const v2f*)(XlL + kt * 4 + s0);

    // S = E * X : 3 output tiles, 12 chained K accumulations each
    v8f acc[3];
#pragma unroll
    for (int nt = 0; nt < 3; ++nt) {
      v8f c = {0.f, 0.f, 0.f, 0.f, 0.f, 0.f, 0.f, 0.f};
#pragma unroll
      for (int kt = 0; kt < 12; ++kt)
        c = __builtin_amdgcn_wmma_f32_16x16x4_f32(
                false, E[nt][kt], false, Bt[kt], (short)0, c, false, false);
      acc[nt] = c;
    }

    // beta = x_t*log2e + M + log2(S)   (S >= exp(min trans) -> never denormal)
#pragma unroll
    for (int nt = 0; nt < 3; ++nt) {
      const float4 lo = *(const float4*)(xt + nt * 16 + roff);
      const float4 hi = *(const float4*)(xt + nt * 16 + roff + 4);
      al[nt][0] = fmaf(lo.x, LOG2E, M + __builtin_amdgcn_logf(acc[nt][0]));
      al[nt][1] = fmaf(lo.y, LOG2E, M + __builtin_amdgcn_logf(acc[nt][1]));
      al[nt][2] = fmaf(lo.z, LOG2E, M + __builtin_amdgcn_logf(acc[nt][2]));
      al[nt][3] = fmaf(lo.w, LOG2E, M + __builtin_amdgcn_logf(acc[nt][3]));
      al[nt][4] = fmaf(hi.x, LOG2E, M + __builtin_amdgcn_logf(acc[nt][4]));
      al[nt][5] = fmaf(hi.y, LOG2E, M + __builtin_amdgcn_logf(acc[nt][5]));
      al[nt][6] = fmaf(hi.z, LOG2E, M + __builtin_amdgcn_logf(acc[nt][6]));
      al[nt][7] = fmaf(hi.w, LOG2E, M + __builtin_amdgcn_logf(acc[nt][7]));
    }
  }

  // fcc[b] = ln2 * log2sumexp2_n(beta[n,b])
  float M = fmaxf(fmaxf(max8(al[0]), max8(al[1])), max8(al[2]));
  M = fmaxf(M, __shfl_xor(M, 16, 32));
  float s = 0.f;
#pragma unroll
  for (int nt = 0; nt < 3; ++nt)
#pragma unroll
    for (int v = 0; v < 8; ++v)
      s += __builtin_amdgcn_exp2f(al[nt][v] - M);
  s += __shfl_xor(s, 16, 32);
  if (half == 0) fcc[b] = LN2 * (M + __builtin_amdgcn_logf(s));
}

// ---------------------------------------------------------------------------
// Kernel 2: _fac, one wave per batch element. Lane k owns states [7k, 7k+6];
// the alpha[i-1] dependency crosses lanes only once per step (__shfl_up of the
// lane's last state). out[b] = fcc_ws[b] - fac[b].
// ---------------------------------------------------------------------------
__launch_bounds__(32)
__global__ void asg_fac_kernel(const float* __restrict__ x,      // [B,T,48]
                               const int*   __restrict__ target, // [B,L]
                               const int*   __restrict__ tsize,  // [B]
                               const float* __restrict__ trans,  // [48,48]
                               const float* __restrict__ fcc,    // [B] (ws)
                               float* __restrict__ out,          // [B]
                               int T, int L) {
  const int b    = blockIdx.x;
  const int lane = threadIdx.x;
  __shared__ float af[7 * 32];

  const int  q   = tsize[b] - 1;            // final state index
  const int* tgb = target + (size_t)b * L;

  int   tg[7];
  float st[7], pv[7];
  bool  mk[7];
#pragma unroll
  for (int j = 0; j < 7; ++j) {
    const int s  = lane * 7 + j;
    const int ss = (s < L) ? s : (L - 1);
    tg[j] = tgb[ss];
    st[j] = trans[tg[j] * 48 + tg[j]];                              // self
    pv[j] = (s >= 1 && s < L) ? trans[tg[j] * 48 + tgb[ss - 1]]     // prev
                              : NEGV;
    mk[j] = (s <= q) && (s < L);
  }

  const float* xb = x + (size_t)b * T * 48;
  float a[7];
#pragma unroll
  for (int j = 0; j < 7; ++j) {
    const int s = lane * 7 + j;
    a[j] = (s == 0) ? xb[tg[0]] : NEGV;
  }

  for (int t = 1; t < T; ++t) {
    const float* xt = xb + (size_t)t * 48;
    float carry = __shfl_up(a[6], 1, 32);   // old alpha[7*lane - 1]
    if (lane == 0) carry = NEGV;
#pragma unroll
    for (int j = 6; j >= 0; --j) {          // descending: reads old a[j-1]
      const float ap   = (j == 0) ? carry : a[j - 1];
      const float stay = a[j] + st[j];
      const float move = ap + pv[j];
      const float mx   = fmaxf(stay, move);
      const float mn   = fminf(stay, move);
      // logaddexp: mx + ln2*log2(1 + 2^((mn-mx)*log2e)); arg >= 1, no denorms
      const float e    = __builtin_amdgcn_exp2f((mn - mx) * LOG2E);
      const float lae  = fmaf(LN2, __builtin_amdgcn_logf(1.0f + e), mx);
      const float nv   = xt[tg[j]] + lae;
      a[j] = mk[j] ? nv : NEGV;
    }
  }

#pragma unroll
  for (int j = 0; j < 7; ++j) af[lane * 7 + j] = a[j];
  __syncthreads();
  if (lane == 0) out[b] = fcc[b] - af[q];
}

// ---------------------------------------------------------------------------
extern "C" void kernel_launch(void* const* d_in, const int* in_sizes, int n_in,
                              void* d_out, int out_size, void* d_ws, size_t ws_size,
                              hipStream_t stream) {
  const float* x      = (const float*)d_in[0];  // [B,T,N]
  const int*   target = (const int*)  d_in[1];  // [B,L]
  const int*   tsize  = (const int*)  d_in[2];  // [B]
  const float* trans  = (const float*)d_in[3];  // [N,N]

  const int B = in_sizes[2];
  const int N = 48;                              // layout specialized to N=48
  const int T = in_sizes[0] / (B * N);
  const int L = in_sizes[1] / B;

  float* fcc_ws = (float*)d_ws;                  // [B] scratch
  float* out    = (float*)d_out;                 // [B]

  asg_fcc_kernel<<<B / 16, 32, 0, stream>>>(x, trans, fcc_ws, T);
  asg_fac_kernel<<<B, 32, 0, stream>>>(x, target, tsize, trans, fcc_ws, out, T, L);
}